// GATNet_47459388621464
// MI455X (gfx1250) — compile-verified
//
#include <hip/hip_runtime.h>

#define N_NODES 50000
#define N_EDGES 400000
#define E_TOT   (N_EDGES + N_NODES)   // with self loops
#define NG      64
#define NEG_SLOPE 0.2f
#define BN_EPS  1e-5f

typedef float v2f __attribute__((ext_vector_type(2)));
typedef float v8f __attribute__((ext_vector_type(8)));

// ---- monotonic float<->uint key for atomicMax-based segment max ------------
__device__ __forceinline__ unsigned fkey(float f) {
  unsigned u = __float_as_uint(f);
  return (u & 0x80000000u) ? ~u : (u | 0x80000000u);
}
__device__ __forceinline__ float keyf(unsigned k) {
  unsigned u = (k & 0x80000000u) ? (k & 0x7fffffffu) : ~k;
  return __uint_as_float(u);
}
__device__ __forceinline__ float lrelu(float x) { return x > 0.f ? x : NEG_SLOPE * x; }

// ---------------------------------------------------------------------------
// fp32 WMMA GEMM: C[M,NN] = A[M,K](lda) @ B[K,NN](row-major, ldb=NN) [+ bias]
// One wave computes a 16x64 strip: 4 accumulators share one A fragment.
// Uses V_WMMA_F32_16X16X4_F32 (full fp32 precision, matches reference math).
// A 16x4 layout: lane(l&15)=row M, (l>>4)*2=K pair; a={A[m][k],A[m][k+1]}.
// B 4x16 layout (symmetric): lane(l&15)=col N, b={B[k][n],B[k+1][n]}.
// C/D: VGPR r -> row (l>>4)*8 + r, col (l&15).
// ---------------------------------------------------------------------------
__global__ void k_wmma_gemm(const float* __restrict__ A, const float* __restrict__ B,
                            const float* __restrict__ bias, float* __restrict__ C,
                            int M, int K, int NN, int lda, int ldc) {
  int wid  = (blockIdx.x * blockDim.x + threadIdx.x) >> 5;
  int lane = threadIdx.x & 31;
  int ntiles = NN >> 6;                 // number of 64-column groups
  int mt = wid / ntiles;
  int nt = wid - mt * ntiles;
  if (mt * 16 >= M) return;             // uniform per wave -> EXEC stays all-1s

  int mrow = mt * 16 + (lane & 15);
  int koff = (lane >> 4) << 1;
  int cb   = (nt << 6) + (lane & 15);

  v8f acc0 = {0.f,0.f,0.f,0.f,0.f,0.f,0.f,0.f};
  v8f acc1 = acc0, acc2 = acc0, acc3 = acc0;

  const float* Arow = A + (size_t)mrow * lda + koff;
  for (int k = 0; k < K; k += 4) {
    v2f a = *(const v2f*)(Arow + k);
    const float* Bp = B + (size_t)(k + koff) * NN + cb;
    v2f b0 = { Bp[0],  Bp[NN] };
    v2f b1 = { Bp[16], Bp[NN + 16] };
    v2f b2 = { Bp[32], Bp[NN + 32] };
    v2f b3 = { Bp[48], Bp[NN + 48] };
    acc0 = __builtin_amdgcn_wmma_f32_16x16x4_f32(false, a, false, b0, (short)0, acc0, false, false);
    acc1 = __builtin_amdgcn_wmma_f32_16x16x4_f32(false, a, false, b1, (short)0, acc1, false, false);
    acc2 = __builtin_amdgcn_wmma_f32_16x16x4_f32(false, a, false, b2, (short)0, acc2, false, false);
    acc3 = __builtin_amdgcn_wmma_f32_16x16x4_f32(false, a, false, b3, (short)0, acc3, false, false);
  }

  float bv0 = 0.f, bv1 = 0.f, bv2 = 0.f, bv3 = 0.f;
  if (bias) { bv0 = bias[cb]; bv1 = bias[cb + 16]; bv2 = bias[cb + 32]; bv3 = bias[cb + 48]; }
  int rbase = (lane >> 4) << 3;
#pragma unroll
  for (int r = 0; r < 8; ++r) {
    float* Cp = C + (size_t)(mt * 16 + rbase + r) * ldc + cb;
    Cp[0]  = acc0[r] + bv0;
    Cp[16] = acc1[r] + bv1;
    Cp[32] = acc2[r] + bv2;
    Cp[48] = acc3[r] + bv3;
  }
}

// ---------------------------------------------------------------------------
// Per-node attention scores: as[n,h] = sum_c xh[n,h,c]*att_src[h,c] (+dst).
// Wave per node; lane covers 8 contiguous channels (head = lane/8); fully
// coalesced 1KB load per wave, then 8-lane xor-shuffle reduction.
// ---------------------------------------------------------------------------
__global__ void k_att_scores(const float* __restrict__ xh,
                             const float* __restrict__ asrc,
                             const float* __restrict__ adst,
                             float* __restrict__ as_, float* __restrict__ ad_) {
  int wid  = (blockIdx.x * blockDim.x + threadIdx.x) >> 5;
  int lane = threadIdx.x & 31;
  if (wid >= N_NODES) return;
  const float* xp = xh + (size_t)wid * 256 + lane * 8;
  float sa = 0.f, sd = 0.f;
#pragma unroll
  for (int j = 0; j < 8; ++j) {
    float v = xp[j];
    sa += v * asrc[lane * 8 + j];
    sd += v * adst[lane * 8 + j];
  }
#pragma unroll
  for (int off = 1; off < 8; off <<= 1) {
    sa += __shfl_xor(sa, off, 32);
    sd += __shfl_xor(sd, off, 32);
  }
  if ((lane & 7) == 0) {
    int h = lane >> 3;
    as_[wid * 4 + h] = sa;
    ad_[wid * 4 + h] = sd;
  }
}

// ---- pass 1: per-(dst,head) max of leaky_relu(as[src]+ad[dst]) -------------
__global__ void k_edge_max(const int* __restrict__ ei, const float* __restrict__ as_,
                           const float* __restrict__ ad_, unsigned* __restrict__ mkey) {
  int e = blockIdx.x * blockDim.x + threadIdx.x;
  if (e >= E_TOT) return;
  int s, d;
  if (e < N_EDGES) { s = ei[e]; d = ei[N_EDGES + e]; } else { s = d = e - N_EDGES; }
  float4 a = *(const float4*)(as_ + 4 * (size_t)s);
  float4 b = *(const float4*)(ad_ + 4 * (size_t)d);
  atomicMax(&mkey[4 * d + 0], fkey(lrelu(a.x + b.x)));
  atomicMax(&mkey[4 * d + 1], fkey(lrelu(a.y + b.y)));
  atomicMax(&mkey[4 * d + 2], fkey(lrelu(a.z + b.z)));
  atomicMax(&mkey[4 * d + 3], fkey(lrelu(a.w + b.w)));
}

// ---- pass 2: e=exp(logit-max); accumulate sum(e) and unnormalized out ------
// Wave per edge; lane handles 8 contiguous channels of its head.
__global__ void k_edge_accum(const int* __restrict__ ei, const float* __restrict__ as_,
                             const float* __restrict__ ad_, const unsigned* __restrict__ mkey,
                             float* __restrict__ ssum, const float* __restrict__ xh,
                             float* __restrict__ oacc) {
  int wid  = (blockIdx.x * blockDim.x + threadIdx.x) >> 5;
  int lane = threadIdx.x & 31;
  if (wid >= E_TOT) return;
  int s, d;
  if (wid < N_EDGES) { s = ei[wid]; d = ei[N_EDGES + wid]; } else { s = d = wid - N_EDGES; }
  int h = lane >> 3;
  float logit = lrelu(as_[4 * (size_t)s + h] + ad_[4 * (size_t)d + h]);
  float ew = expf(logit - keyf(mkey[4 * (size_t)d + h]));
  if ((lane & 7) == 0) atomicAdd(&ssum[4 * (size_t)d + h], ew);
  const float* xp = xh + (size_t)s * 256 + lane * 8;
  float*       op = oacc + (size_t)d * 256 + lane * 8;
  float4 v0 = *(const float4*)(xp);
  float4 v1 = *(const float4*)(xp + 4);
  atomicAdd(op + 0, ew * v0.x); atomicAdd(op + 1, ew * v0.y);
  atomicAdd(op + 2, ew * v0.z); atomicAdd(op + 3, ew * v0.w);
  atomicAdd(op + 4, ew * v1.x); atomicAdd(op + 5, ew * v1.y);
  atomicAdd(op + 6, ew * v1.z); atomicAdd(op + 7, ew * v1.w);
}

// ---- layer 0 epilogue a: normalize by sum(e) and add conv bias -------------
__global__ void k_norm_bias(float* __restrict__ oacc, const float* __restrict__ ssum,
                            const float* __restrict__ bias) {
  int i = blockIdx.x * blockDim.x + threadIdx.x;
  if (i >= N_NODES * 256) return;
  int n = i >> 8, c = i & 255;
  oacc[i] = oacc[i] / ssum[n * 4 + (c >> 6)] + bias[c];
}

// ---- layer 0 epilogue b: in-place BN + ELU on hcat slice (ld=192) ----------
__global__ void k_bn_elu0(float* __restrict__ h, const float* __restrict__ g,
                          const float* __restrict__ b, const float* __restrict__ m,
                          const float* __restrict__ v) {
  int i = blockIdx.x * blockDim.x + threadIdx.x;
  if (i >= N_NODES * 64) return;
  int n = i >> 6, c = i & 63;
  float x = h[(size_t)n * 192 + c];
  x = (x - m[c]) * rsqrtf(v[c] + BN_EPS) * g[c] + b[c];
  h[(size_t)n * 192 + c] = x > 0.f ? x : expm1f(x);
}

// ---- layers 1/2 epilogue: normalize, head-mean, +bias, BN, ELU, residual ---
__global__ void k_layer_epi(const float* __restrict__ oacc, const float* __restrict__ ssum,
                            const float* __restrict__ bias, const float* __restrict__ g,
                            const float* __restrict__ bb, const float* __restrict__ m,
                            const float* __restrict__ v, const float* __restrict__ prev,
                            float* __restrict__ outp) {
  int i = blockIdx.x * blockDim.x + threadIdx.x;
  if (i >= N_NODES * 64) return;
  int n = i >> 6, c = i & 63;
  float acc = 0.f;
#pragma unroll
  for (int h = 0; h < 4; ++h)
    acc += oacc[(size_t)n * 256 + h * 64 + c] / ssum[n * 4 + h];
  float x = acc * 0.25f + bias[c];
  x = (x - m[c]) * rsqrtf(v[c] + BN_EPS) * g[c] + bb[c];
  x = x > 0.f ? x : expm1f(x);
  outp[(size_t)n * 192 + c] = x + prev[(size_t)n * 192 + c];
}

// ---- pooling: score = hjk . att_W + att_b; per-graph max via key atomic ----
__global__ void k_pool_score(const float* __restrict__ hjk, const float* __restrict__ attW,
                             const float* __restrict__ attb, const int* __restrict__ batch,
                             float* __restrict__ score, unsigned* __restrict__ gmkey) {
  int wid  = (blockIdx.x * blockDim.x + threadIdx.x) >> 5;
  int lane = threadIdx.x & 31;
  if (wid >= N_NODES) return;
  float2 hv = *(const float2*)(hjk + (size_t)wid * 64 + lane * 2);
  float2 wv = *(const float2*)(attW + lane * 2);
  float p = hv.x * wv.x + hv.y * wv.y;
#pragma unroll
  for (int off = 1; off < 32; off <<= 1) p += __shfl_xor(p, off, 32);
  if (lane == 0) {
    float sc = p + attb[0];
    score[wid] = sc;
    atomicMax(&gmkey[batch[wid]], fkey(sc));
  }
}

__global__ void k_pool_exp(float* __restrict__ score, const int* __restrict__ batch,
                           const unsigned* __restrict__ gmkey, float* __restrict__ gsum) {
  int n = blockIdx.x * blockDim.x + threadIdx.x;
  if (n >= N_NODES) return;
  int b = batch[n];
  float e = expf(score[n] - keyf(gmkey[b]));
  score[n] = e;
  atomicAdd(&gsum[b], e);
}

__global__ void k_pool_accum(const float* __restrict__ hjk, const float* __restrict__ score,
                             const int* __restrict__ batch, const float* __restrict__ gsum,
                             float* __restrict__ hg) {
  int wid  = (blockIdx.x * blockDim.x + threadIdx.x) >> 5;
  int lane = threadIdx.x & 31;
  if (wid >= N_NODES) return;
  int b = batch[wid];
  float a = score[wid] / gsum[b];
  float2 hv = *(const float2*)(hjk + (size_t)wid * 64 + lane * 2);
  atomicAdd(&hg[b * 64 + lane * 2 + 0], a * hv.x);
  atomicAdd(&hg[b * 64 + lane * 2 + 1], a * hv.y);
}

// ---- tiny classifier --------------------------------------------------------
__global__ void k_cls1(const float* __restrict__ hg, const float* __restrict__ W1,
                       const float* __restrict__ b1, float* __restrict__ hidden) {
  int i = blockIdx.x * blockDim.x + threadIdx.x;
  if (i >= NG * 32) return;
  int g = i >> 5, j = i & 31;
  float acc = b1[j];
  for (int c = 0; c < 64; ++c) acc += hg[g * 64 + c] * W1[c * 32 + j];
  hidden[i] = acc > 0.f ? acc : 0.f;
}

__global__ void k_cls2(const float* __restrict__ hidden, const float* __restrict__ W2,
                       const float* __restrict__ b2, float* __restrict__ out) {
  int i = blockIdx.x * blockDim.x + threadIdx.x;
  if (i >= NG * 2) return;
  int g = i >> 1, j = i & 1;
  float acc = b2[j];
  for (int c = 0; c < 32; ++c) acc += hidden[g * 32 + c] * W2[c * 2 + j];
  out[i] = acc;
}

// ===========================================================================
extern "C" void kernel_launch(void* const* d_in, const int* in_sizes, int n_in,
                              void* d_out, int out_size, void* d_ws, size_t ws_size,
                              hipStream_t stream) {
  (void)in_sizes; (void)n_in; (void)out_size; (void)ws_size;
  const float* x        = (const float*)d_in[0];
  const int*   ei       = (const int*)d_in[1];
  const int*   batch    = (const int*)d_in[2];
  const float* conv0_W  = (const float*)d_in[3];
  const float* conv0_as = (const float*)d_in[4];
  const float* conv0_ad = (const float*)d_in[5];
  const float* conv0_b  = (const float*)d_in[6];
  const float* pre0_W   = (const float*)d_in[7];
  const float* pre0_b   = (const float*)d_in[8];
  const float* convs_W  = (const float*)d_in[9];
  const float* convs_as = (const float*)d_in[10];
  const float* convs_ad = (const float*)d_in[11];
  const float* convs_b  = (const float*)d_in[12];
  const float* bng      = (const float*)d_in[13];
  const float* bnb      = (const float*)d_in[14];
  const float* bnm      = (const float*)d_in[15];
  const float* bnv      = (const float*)d_in[16];
  const float* jump_W   = (const float*)d_in[17];
  const float* jump_b   = (const float*)d_in[18];
  const float* att_W    = (const float*)d_in[19];
  const float* att_b    = (const float*)d_in[20];
  const float* cls_W1   = (const float*)d_in[21];
  const float* cls_b1   = (const float*)d_in[22];
  const float* cls_W2   = (const float*)d_in[23];
  const float* cls_b2   = (const float*)d_in[24];
  float* out = (float*)d_out;

  const size_t N = N_NODES;
  float*    ws     = (float*)d_ws;
  float*    xh     = ws;                       // [N,256]
  float*    oacc   = xh + N * 256;             // [N,256]
  float*    hcat   = oacc + N * 256;           // [N,192] (rep0|rep1|rep2)
  float*    as_    = hcat + N * 192;           // [N,4]
  float*    ad_    = as_ + N * 4;              // [N,4]
  unsigned* mkey   = (unsigned*)(ad_ + N * 4); // [N,4]
  float*    ssum   = (float*)(mkey + N * 4);   // [N,4]
  float*    score  = ssum + N * 4;             // [N]
  unsigned* gmkey  = (unsigned*)(score + N);   // [G]
  float*    gsum   = (float*)(gmkey + NG);     // [G]
  float*    hg     = gsum + NG;                // [G,64]
  float*    hidden = hg + NG * 64;             // [G,32]
  float*    hjk    = xh;                       // alias: xh is dead before JK

  const int EB_T = (E_TOT + 255) / 256;        // thread-per-edge blocks
  const int EB_W = (E_TOT + 7) / 8;            // wave-per-edge blocks
  const int NB_W = (N_NODES + 7) / 8;          // wave-per-node blocks

  auto gemm = [&](const float* A, const float* B, const float* bias, float* C,
                  int M, int K, int NN, int lda, int ldc) {
    int waves = ((M + 15) / 16) * (NN / 64);
    k_wmma_gemm<<<(waves + 7) / 8, 256, 0, stream>>>(A, B, bias, C, M, K, NN, lda, ldc);
  };
  auto gat_edges = [&](const float* att_s, const float* att_d) {
    k_att_scores<<<NB_W, 256, 0, stream>>>(xh, att_s, att_d, as_, ad_);
    hipMemsetAsync(mkey, 0, N * 4 * sizeof(unsigned), stream); // key 0 == -inf
    hipMemsetAsync(ssum, 0, N * 4 * sizeof(float), stream);
    hipMemsetAsync(oacc, 0, N * 256 * sizeof(float), stream);
    k_edge_max<<<EB_T, 256, 0, stream>>>(ei, as_, ad_, mkey);
    k_edge_accum<<<EB_W, 256, 0, stream>>>(ei, as_, ad_, mkey, ssum, xh, oacc);
  };

  // ---- layer 0: GAT(concat) -> pre-proj -> BN -> ELU -----------------------
  gemm(x, conv0_W, nullptr, xh, N_NODES, 128, 256, 128, 256);
  gat_edges(conv0_as, conv0_ad);
  k_norm_bias<<<(N_NODES * 256 + 255) / 256, 256, 0, stream>>>(oacc, ssum, conv0_b);
  gemm(oacc, pre0_W, pre0_b, hcat, N_NODES, 256, 64, 256, 192);
  k_bn_elu0<<<(N_NODES * 64 + 255) / 256, 256, 0, stream>>>(hcat, bng, bnb, bnm, bnv);

  // ---- layers 1..2: GAT(mean) -> BN -> ELU -> residual ---------------------
  for (int i = 0; i < 2; ++i) {
    gemm(hcat + i * 64, convs_W + (size_t)i * 64 * 256, nullptr, xh,
         N_NODES, 64, 256, 192, 256);
    gat_edges(convs_as + i * 256, convs_ad + i * 256);
    k_layer_epi<<<(N_NODES * 64 + 255) / 256, 256, 0, stream>>>(
        oacc, ssum, convs_b + i * 64,
        bng + (i + 1) * 64, bnb + (i + 1) * 64, bnm + (i + 1) * 64, bnv + (i + 1) * 64,
        hcat + i * 64, hcat + (i + 1) * 64);
  }

  // ---- jumping knowledge: [N,192] @ [192,64] + b ---------------------------
  gemm(hcat, jump_W, jump_b, hjk, N_NODES, 192, 64, 192, 64);

  // ---- attentional pooling over graphs -------------------------------------
  hipMemsetAsync(gmkey, 0, NG * sizeof(unsigned), stream);
  hipMemsetAsync(gsum, 0, NG * sizeof(float), stream);
  hipMemsetAsync(hg, 0, NG * 64 * sizeof(float), stream);
  k_pool_score<<<NB_W, 256, 0, stream>>>(hjk, att_W, att_b, batch, score, gmkey);
  k_pool_exp<<<(N_NODES + 255) / 256, 256, 0, stream>>>(score, batch, gmkey, gsum);
  k_pool_accum<<<NB_W, 256, 0, stream>>>(hjk, score, batch, gsum, hg);

  // ---- classifier ----------------------------------------------------------
  k_cls1<<<(NG * 32 + 255) / 256, 256, 0, stream>>>(hg, cls_W1, cls_b1, hidden);
  k_cls2<<<1, 256, 0, stream>>>(hidden, cls_W2, cls_b2, out);
}